// GraphDiffusion_63299228008754
// MI455X (gfx1250) — compile-verified
//
#include <hip/hip_runtime.h>
#include <hip/hip_bf16.h>
#include <stddef.h>

#define N_NODES 100000
#define N_EDGES 1000000
#define NUM_GRAPHS 500
#define EPSV 1e-5f

typedef __attribute__((ext_vector_type(2))) float v2f;
typedef __attribute__((ext_vector_type(8))) float v8f;

// ---------------------------------------------------------------- utilities
__global__ void zero_f32_kernel(float* __restrict__ p, int n) {
    int i = blockIdx.x * blockDim.x + threadIdx.x;
    if (i < n) p[i] = 0.0f;
}

// ------------------------------------------------------------------ degrees
__global__ void degree_kernel(const int* __restrict__ dst, float* __restrict__ deg, int n_edges) {
    int e = blockIdx.x * blockDim.x + threadIdx.x;
    if (e < n_edges) atomicAdd(&deg[dst[e]], 1.0f);
}

__global__ void dis_kernel(const float* __restrict__ deg, float* __restrict__ dis, int n) {
    int i = blockIdx.x * blockDim.x + threadIdx.x;
    if (i < n) dis[i] = rsqrtf(deg[i] + 1.0f);   // in-degree + self-loop
}

// ------------------------------------------------------- WMMA fp32 GEMM
// H[M,N] = X[M,K] @ W[K,N].  M multiple of 16. One 16-row M tile (16xN outputs) per wave,
// NT = N/16 accumulators per wave (A-fragment reused across the N tiles).
// V_WMMA_F32_16X16X4_F32 layouts (ISA 7.12.2):
//   A (16x4): lane L<16 row M=L holds K=0(v0),K=1(v1); lane 16+L holds K=2,K=3
//   B (4x16): lane L<16 col N=L holds K=0(v0),K=1(v1); lane 16+L holds K=2,K=3
//   C/D: VGPR r, lanes 0-15 -> (M=r, N=lane); lanes 16-31 -> (M=r+8, N=lane-16)
//
// W is staged in LDS PRE-PAIRED along K: lw2[(k>>1)*N + n] = {W[k][n], W[k+1][n]}
// so every B fragment is ONE aligned ds_load_b64 directly into an even VGPR
// pair (no v_dual_mov repacking before each WMMA).
template <int K, int N>
__global__ void gemm_wmma_f32_kernel(const float* __restrict__ X,
                                     const float* __restrict__ W,
                                     float* __restrict__ H, int M) {
    __shared__ float2 lw2[(K / 2) * N];
    const int tid = threadIdx.x;
    {
        float* lws = (float*)lw2;
        for (int i = tid; i < K * N; i += blockDim.x) {
            const int k = i / N;
            const int n = i % N;
            // element W[k][n] goes to pair (k>>1), slot (k&1)
            lws[(size_t)(k >> 1) * (2 * N) + n * 2 + (k & 1)] = W[i];
        }
    }
    __syncthreads();

    const int wave  = tid >> 5;
    const int lane  = tid & 31;
    const int tileM = blockIdx.x * 8 + wave;
    const int numTiles = M >> 4;
    if (tileM >= numTiles) return;          // wave-uniform exit, EXEC stays all-1s

    const int mrow  = (tileM << 4) + (lane & 15);
    const int khalf = (lane >> 4) << 1;     // 0 for lanes 0-15, 2 for lanes 16-31
    const int ncol  = lane & 15;
    constexpr int NT = N / 16;

    v8f acc[NT];
#pragma unroll
    for (int t = 0; t < NT; ++t) acc[t] = (v8f)0.0f;

    const float* xrow = X + (size_t)mrow * K;
    const float2* lwrow = lw2 + (khalf >> 1) * N + ncol;   // this lane's column/half
    for (int k0 = 0; k0 < K; k0 += 4) {
        v2f a;
        a.x = xrow[k0 + khalf + 0];
        a.y = xrow[k0 + khalf + 1];
#pragma unroll
        for (int t = 0; t < NT; ++t) {
            const float2 bp = lwrow[(k0 >> 1) * N + t * 16];  // single ds_load_b64
            v2f b;
            b.x = bp.x;
            b.y = bp.y;
            acc[t] = __builtin_amdgcn_wmma_f32_16x16x4_f32(
                false, a, false, b, (short)0, acc[t], false, false);
        }
    }

    const int rbase = (tileM << 4) + ((lane >> 4) << 3);   // +8 for upper half-wave
#pragma unroll
    for (int r = 0; r < 8; ++r) {
        float* orow = H + (size_t)(rbase + r) * N;
#pragma unroll
        for (int t = 0; t < NT; ++t) orow[t * 16 + ncol] = acc[t][r];
    }
}

// --------------------------------------------- self-loop init: agg = h*dis^2
template <int F, int SH>
__global__ void self_init_kernel(const float* __restrict__ h, const float* __restrict__ dis,
                                 float* __restrict__ agg, int total) {
    int idx = blockIdx.x * blockDim.x + threadIdx.x;
    if (idx >= total) return;
    int node = idx >> SH;
    float d = dis[node];
    agg[idx] = h[idx] * d * d;
}

// --------------------------- edge scatter: agg[dst] += h[src]*dis[s]*dis[d]
// LPE lanes (float4 each) cooperate on one edge -> coalesced B128 gather,
// non-returning global_atomic_add_f32 scatter (resolves in the 192MB L2).
template <int F, int LPE>
__global__ void edge_scatter_kernel(const int* __restrict__ src, const int* __restrict__ dst,
                                    const float* __restrict__ dis,
                                    const float* __restrict__ h, float* __restrict__ agg,
                                    int n_edges) {
    int tid = blockIdx.x * blockDim.x + threadIdx.x;
    int e = tid / LPE;
    int l = tid % LPE;
    if (e >= n_edges) return;
    int s = src[e], d = dst[e];
    float c = dis[s] * dis[d];
    const float4 hv = *(const float4*)(h + (size_t)s * F + l * 4);
    float* ap = agg + (size_t)d * F + l * 4;
    atomicAdd(ap + 0, hv.x * c);
    atomicAdd(ap + 1, hv.y * c);
    atomicAdd(ap + 2, hv.z * c);
    atomicAdd(ap + 3, hv.w * c);
}

// --------------------------------------- fused  (+bias) -> BN(eval) -> ReLU
template <int F, int SH>
__global__ void bn_relu_kernel(const float* __restrict__ agg,
                               const float* __restrict__ b, const float* __restrict__ g,
                               const float* __restrict__ be, const float* __restrict__ m,
                               const float* __restrict__ v, float* __restrict__ out, int total) {
    int idx = blockIdx.x * blockDim.x + threadIdx.x;
    if (idx >= total) return;
    int f = idx & (F - 1);
    float val = agg[idx] + b[f];
    float y = (val - m[f]) * rsqrtf(v[f] + EPSV) * g[f] + be[f];
    out[idx] = fmaxf(y, 0.0f);
}

// --------------------------------------------------------- global mean pool
__global__ void pool_kernel(const float* __restrict__ h, const int* __restrict__ batch,
                            float* __restrict__ sums, float* __restrict__ cnt, int n_nodes) {
    int tid = blockIdx.x * blockDim.x + threadIdx.x;
    int node = tid >> 5;
    int lane = tid & 31;
    if (node >= n_nodes) return;
    int gr = batch[node];
    atomicAdd(&sums[gr * 32 + lane], h[(size_t)node * 32 + lane]);
    if (lane == 0) atomicAdd(&cnt[gr], 1.0f);
}

// -------------------------------------------------------- classifier MLP
__global__ void classifier_kernel(const float* __restrict__ sums, const float* __restrict__ cnt,
                                  const float* __restrict__ Wc1, const float* __restrict__ bc1,
                                  const float* __restrict__ Wc2, const float* __restrict__ bc2,
                                  float* __restrict__ out, int n_graphs) {
    int gidx = blockIdx.x * blockDim.x + threadIdx.x;
    if (gidx >= n_graphs) return;
    float inv = 1.0f / fmaxf(cnt[gidx], 1.0f);
    float z[32];
#pragma unroll
    for (int f = 0; f < 32; ++f) z[f] = sums[gidx * 32 + f] * inv;
    float hid[16];
#pragma unroll
    for (int j = 0; j < 16; ++j) {
        float acc = bc1[j];
        for (int f = 0; f < 32; ++f) acc += z[f] * Wc1[f * 16 + j];
        hid[j] = fmaxf(acc, 0.0f);
    }
#pragma unroll
    for (int c = 0; c < 2; ++c) {
        float acc = bc2[c];
        for (int j = 0; j < 16; ++j) acc += hid[j] * Wc2[j * 2 + c];
        out[gidx * 2 + c] = acc;
    }
}

extern "C" void kernel_launch(void* const* d_in, const int* in_sizes, int n_in,
                              void* d_out, int out_size, void* d_ws, size_t ws_size,
                              hipStream_t stream) {
    const float* x     = (const float*)d_in[0];
    const int*   ei    = (const int*)d_in[1];
    const int*   batch = (const int*)d_in[2];
    const float* W1  = (const float*)d_in[3];
    const float* b1  = (const float*)d_in[4];
    const float* g1  = (const float*)d_in[5];
    const float* be1 = (const float*)d_in[6];
    const float* m1  = (const float*)d_in[7];
    const float* v1  = (const float*)d_in[8];
    const float* W2  = (const float*)d_in[9];
    const float* b2  = (const float*)d_in[10];
    const float* g2  = (const float*)d_in[11];
    const float* be2 = (const float*)d_in[12];
    const float* m2  = (const float*)d_in[13];
    const float* v2  = (const float*)d_in[14];
    const float* Wc1 = (const float*)d_in[15];
    const float* bc1 = (const float*)d_in[16];
    const float* Wc2 = (const float*)d_in[17];
    const float* bc2 = (const float*)d_in[18];
    float* out = (float*)d_out;

    const int N = N_NODES, E = N_EDGES, G = NUM_GRAPHS;
    const int* src = ei;
    const int* dst = ei + E;

    // workspace layout (floats); bufA reused for h2/agg2 after layer 1
    float* ws   = (float*)d_ws;
    float* deg  = ws;                         // N
    float* dis  = ws + N;                     // N
    float* bufA = ws + 2 * (size_t)N;         // N*64  (h1, later h2 + agg2)
    float* bufB = bufA + (size_t)N * 64;      // N*64  (agg1 -> relu-h1)
    float* h2   = bufA;                       // N*32
    float* agg2 = bufA + (size_t)N * 32;      // N*32
    float* sums = bufB + (size_t)N * 64;      // G*32
    float* cnt  = sums + (size_t)G * 32;      // G

    const int TB = 256;

    // 1) degrees and symmetric-norm coefficients
    zero_f32_kernel<<<(N + TB - 1) / TB, TB, 0, stream>>>(deg, N);
    degree_kernel<<<(E + TB - 1) / TB, TB, 0, stream>>>(dst, deg, E);
    dis_kernel<<<(N + TB - 1) / TB, TB, 0, stream>>>(deg, dis, N);

    // 2) layer 1: h1 = x @ W1   (WMMA fp32, 6250 M-tiles, 8 waves/block)
    gemm_wmma_f32_kernel<64, 64><<<(N / 16 + 7) / 8, TB, 0, stream>>>(x, W1, bufA, N);
    self_init_kernel<64, 6><<<(N * 64 + TB - 1) / TB, TB, 0, stream>>>(bufA, dis, bufB, N * 64);
    edge_scatter_kernel<64, 16><<<((long long)E * 16 + TB - 1) / TB, TB, 0, stream>>>(
        src, dst, dis, bufA, bufB, E);
    bn_relu_kernel<64, 6><<<(N * 64 + TB - 1) / TB, TB, 0, stream>>>(
        bufB, b1, g1, be1, m1, v1, bufB, N * 64);

    // 3) layer 2: h2 = relu_h1 @ W2
    gemm_wmma_f32_kernel<64, 32><<<(N / 16 + 7) / 8, TB, 0, stream>>>(bufB, W2, h2, N);
    self_init_kernel<32, 5><<<(N * 32 + TB - 1) / TB, TB, 0, stream>>>(h2, dis, agg2, N * 32);
    edge_scatter_kernel<32, 8><<<((long long)E * 8 + TB - 1) / TB, TB, 0, stream>>>(
        src, dst, dis, h2, agg2, E);
    bn_relu_kernel<32, 5><<<(N * 32 + TB - 1) / TB, TB, 0, stream>>>(
        agg2, b2, g2, be2, m2, v2, agg2, N * 32);

    // 4) global mean pool + classifier
    zero_f32_kernel<<<(G * 32 + G + TB - 1) / TB, TB, 0, stream>>>(sums, G * 32 + G);
    pool_kernel<<<((long long)N * 32 + TB - 1) / TB, TB, 0, stream>>>(agg2, batch, sums, cnt, N);
    classifier_kernel<<<(G + TB - 1) / TB, TB, 0, stream>>>(sums, cnt, Wc1, bc1, Wc2, bc2, out, G);
}